// Model_22265110462495
// MI455X (gfx1250) — compile-verified
//
#include <hip/hip_runtime.h>

// out[i,:] = var[i,:] + sum_{j : sorted_indices[j]==i} value[j,:]
//
// MI455X (gfx1250) reasoning:
//  - HBM-bound stream: ~2.05 GB moved, ~88 us floor @ 23.3 TB/s. No matrix
//    content -> no WMMA; optimize the CDNA5 memory path instead.
//  - wave32: one wave per output row, lane l covers columns [4l, 4l+4) ->
//    one global_load_b128 per lane covers the 512B row, fully coalesced.
//  - Lane-parallel lower_bound: wave 0 lanes 0..7 binary-search the block's
//    8 rows simultaneously (search loop trip count is ~log2(M), at most one
//    apart across lanes), results shared via LDS + s_barrier. 8x fewer probe
//    loads than one-search-per-wave, and the probe latency is paid once.
//  - var/value/out are touched exactly once: non-temporal (TH_NT) loads and
//    stores keep the 2 GB stream from evicting the hot 8 MB index array in
//    the 192 MB L2, which the searches and segment scans hammer.
//  - __builtin_prefetch -> global_prefetch_b8 covers the idx[j] -> value[j]
//    dependent-load chain inside each segment scan.

typedef float v4f __attribute__((ext_vector_type(4)));

#define DCOLS 128
#define RPB 8   // rows per block: 256 threads = 8 wave32 = 8 rows

__global__ __launch_bounds__(256) void segadd_kernel(
    const float* __restrict__ var,
    const float* __restrict__ value,
    const int*   __restrict__ idx,     // sorted, length M, values in [0, N)
    float*       __restrict__ out,
    int N, int M)
{
    __shared__ int s_lo[RPB];

    const int lane = threadIdx.x & 31;
    const int wave = threadIdx.x >> 5;
    const int base = (int)blockIdx.x * RPB;

    // --- Lane-parallel lower_bound for the whole block (wave 0, lanes 0..7).
    if (wave == 0 && lane < RPB) {
        int target = base + lane;
        if (target >= N) target = N - 1;     // harmless clamp for tail block
        int lo = 0, cnt = M;
        while (cnt > 0) {                    // trip count ~log2(M), may differ by one per lane
            int half  = cnt >> 1;
            int probe = lo + half;
            if (idx[probe] < target) { lo = probe + 1; cnt -= half + 1; }
            else                     { cnt = half; }
        }
        s_lo[lane] = lo;
    }
    __syncthreads();

    const int r = base + wave;
    if (r >= N) return;

    const long long col = (long long)lane * 4;

    // Streamed read-once data: non-temporal to protect L2 for idx[].
    v4f acc = __builtin_nontemporal_load(
        (const v4f*)(var + (long long)r * DCOLS + col));

    long long j = s_lo[wave];
    if (j < M)     __builtin_prefetch(value + j * (long long)DCOLS + col, 0, 0);
    if (j + 1 < M) __builtin_prefetch(value + (j + 1) * (long long)DCOLS + col, 0, 0);

    // Forward scan of this row's segment (avg length M/N = 2), in index
    // order -> bit-identical to jax segment_sum, fully deterministic.
    while (j < M && idx[j] == r) {
        if (j + 2 < M)
            __builtin_prefetch(value + (j + 2) * (long long)DCOLS + col, 0, 0);
        v4f v = __builtin_nontemporal_load(
            (const v4f*)(value + j * (long long)DCOLS + col));
        acc += v;
        ++j;
    }

    __builtin_nontemporal_store(acc, (v4f*)(out + (long long)r * DCOLS + col));
}

extern "C" void kernel_launch(void* const* d_in, const int* in_sizes, int n_in,
                              void* d_out, int out_size, void* d_ws, size_t ws_size,
                              hipStream_t stream)
{
    const float* var            = (const float*)d_in[0];
    const float* value          = (const float*)d_in[1];
    const int*   sorted_indices = (const int*)d_in[2];
    // d_in[3] = pos : unused by the reference computation
    float* out = (float*)d_out;

    const int M = in_sizes[2];          // number of scatter indices (2,000,000)
    const int N = in_sizes[0] / DCOLS;  // rows of var/out        (1,000,000)

    dim3 block(256);
    dim3 grid((unsigned)((N + RPB - 1) / RPB));
    segadd_kernel<<<grid, block, 0, stream>>>(var, value, sorted_indices, out, N, M);
}